// ControlledChaoticLayer_71683004170589
// MI455X (gfx1250) — compile-verified
//
#include <hip/hip_runtime.h>
#include <hip/hip_bf16.h>

typedef __attribute__((ext_vector_type(2))) float v2f;
typedef __attribute__((ext_vector_type(8))) float v8f;

#define B_ 64
#define D_ 64
#define S_ 4096
#define H_ 128
#define CS_ 0.1f
#define EPS_ 1e-5f

// workspace layout (bytes)
#define WS_CT   0u                 // (B,S,3) f32 = 3,145,728
#define WS_ST   3145728u           // (S,B,3) f32 = 3,145,728
#define WS_CI0  6291456u           // (B,3)  f32 = 768
#define WS_MOM  6292224u           // 9 doubles = 72
#define WS_SC   6292304u           // (H) f32 = 512
#define WS_SH   6292816u           // (H) f32 = 512

// ---------------------------------------------------------------------------
// Kernel 1: ci = x^T Wi + bi ; ct = CS * ci Cm^T ; keep ci at s==0 as init.
// One thread per (b,s). Coalesced over s; Wi/Cm are wave-uniform (s_loads).
// ---------------------------------------------------------------------------
__global__ __launch_bounds__(256) void k_ct(const float* __restrict__ x,
                                            const float* __restrict__ Wi,
                                            const float* __restrict__ bi,
                                            const float* __restrict__ Cm,
                                            float* __restrict__ ct,
                                            float* __restrict__ ci0)
{
    const int b = blockIdx.x >> 4;
    const int s = ((blockIdx.x & 15) << 8) | threadIdx.x;
    const float* xb = x + (size_t)b * D_ * S_ + s;

    float a0 = bi[0], a1 = bi[1], a2 = bi[2];
#pragma unroll 8
    for (int d = 0; d < D_; ++d) {
        float v = xb[(size_t)d * S_];
        const int dp = (d + 8 < D_) ? (d + 8) : (D_ - 1);   // clamped: stay in-bounds
        __builtin_prefetch(&xb[(size_t)dp * S_], 0, 1);     // global_prefetch_b8
        a0 = fmaf(v, Wi[d],            a0);
        a1 = fmaf(v, Wi[D_ + d],       a1);
        a2 = fmaf(v, Wi[2 * D_ + d],   a2);
    }
    float* ctp = ct + ((size_t)b * S_ + s) * 3;
    ctp[0] = CS_ * fmaf(Cm[0], a0, fmaf(Cm[1], a1, Cm[2] * a2));
    ctp[1] = CS_ * fmaf(Cm[3], a0, fmaf(Cm[4], a1, Cm[5] * a2));
    ctp[2] = CS_ * fmaf(Cm[6], a0, fmaf(Cm[7], a1, Cm[8] * a2));
    if (s == 0) { ci0[b * 3 + 0] = a0; ci0[b * 3 + 1] = a1; ci0[b * 3 + 2] = a2; }
}

// ---------------------------------------------------------------------------
// Kernel 2: sequential RK4 (3/8 rule) Lorenz scan. One lane per batch chain,
// state in registers, inner 10-step loop fully unrolled, fmaf-formulated to
// minimize the dependent-chain latency. Also accumulates the 3-vector mean
// and 3x3 second moments of the states in doubles (for analytic BatchNorm).
// ---------------------------------------------------------------------------
__global__ void k_scan(const float* __restrict__ ct, const float* __restrict__ ci0,
                       const float* __restrict__ sig, const float* __restrict__ rho_,
                       const float* __restrict__ bet,
                       float* __restrict__ st, double* __restrict__ mom)
{
    const int b = threadIdx.x;                 // 0..63
    const float sigma = sig[0], rho = rho_[0], beta = bet[0];
    const float h = 0.1f, h3 = h / 3.0f, h8 = h / 8.0f, third = 1.0f / 3.0f;

    float X = ci0[b * 3 + 0], Y = ci0[b * 3 + 1], Z = ci0[b * 3 + 2];
    double sx = 0, sy = 0, sz = 0;
    double sxx = 0, sxy = 0, sxz = 0, syy = 0, syz = 0, szz = 0;

    const float* cp = ct + (size_t)b * S_ * 3;
    for (int s = 0; s < S_; ++s) {
        const float c0 = cp[0], c1 = cp[1], c2 = cp[2];
        cp += 3;
        __builtin_prefetch(cp + 3 * 16, 0, 1);  // stays inside ws (st follows ct)
#pragma unroll
        for (int it = 0; it < 10; ++it) {
            // k1 = f(y)
            float k1x = fmaf(sigma, Y - X, c0);
            float k1y = fmaf(X, rho - Z, c1 - Y);
            float k1z = fmaf(X, Y, fmaf(-beta, Z, c2));
            // k2 = f(y + h/3 * k1)
            float x2 = fmaf(h3, k1x, X), y2 = fmaf(h3, k1y, Y), z2 = fmaf(h3, k1z, Z);
            float k2x = fmaf(sigma, y2 - x2, c0);
            float k2y = fmaf(x2, rho - z2, c1 - y2);
            float k2z = fmaf(x2, y2, fmaf(-beta, z2, c2));
            // k3 = f(y + h*(k2 - k1/3))
            float x3 = fmaf(h, k2x - k1x * third, X);
            float y3 = fmaf(h, k2y - k1y * third, Y);
            float z3 = fmaf(h, k2z - k1z * third, Z);
            float k3x = fmaf(sigma, y3 - x3, c0);
            float k3y = fmaf(x3, rho - z3, c1 - y3);
            float k3z = fmaf(x3, y3, fmaf(-beta, z3, c2));
            // k4 = f(y + h*(k1 - k2 + k3))
            float x4 = fmaf(h, k1x - k2x + k3x, X);
            float y4 = fmaf(h, k1y - k2y + k3y, Y);
            float z4 = fmaf(h, k1z - k2z + k3z, Z);
            float k4x = fmaf(sigma, y4 - x4, c0);
            float k4y = fmaf(x4, rho - z4, c1 - y4);
            float k4z = fmaf(x4, y4, fmaf(-beta, z4, c2));
            // y += h/8 * (k1 + 3*(k2+k3) + k4)
            X = fmaf(h8, fmaf(3.0f, k2x + k3x, k1x + k4x), X);
            Y = fmaf(h8, fmaf(3.0f, k2y + k3y, k1y + k4y), Y);
            Z = fmaf(h8, fmaf(3.0f, k2z + k3z, k1z + k4z), Z);
        }
        float* sp = st + ((size_t)s * B_ + b) * 3;   // (S,B,3): coalesced store
        sp[0] = X; sp[1] = Y; sp[2] = Z;
        sx += X; sy += Y; sz += Z;
        sxx += (double)X * X; sxy += (double)X * Y; sxz += (double)X * Z;
        syy += (double)Y * Y; syz += (double)Y * Z; szz += (double)Z * Z;
    }

    __shared__ double red[B_][9];
    red[b][0] = sx;  red[b][1] = sy;  red[b][2] = sz;
    red[b][3] = sxx; red[b][4] = sxy; red[b][5] = sxz;
    red[b][6] = syy; red[b][7] = syz; red[b][8] = szz;
    __syncthreads();
    if (b < 9) {
        double t = 0.0;
        for (int i = 0; i < B_; ++i) t += red[i][b];
        mom[b] = t;
    }
}

// ---------------------------------------------------------------------------
// Kernel 3: per-h analytic BatchNorm parameters.
// var_h = Wo_h^T Cov Wo_h ; out = Sc[h]*(Wo_h . state) + Sh[h]  (bo cancels).
// ---------------------------------------------------------------------------
__global__ void k_bnparam(const double* __restrict__ mom, const float* __restrict__ Wo,
                          const float* __restrict__ gamma, const float* __restrict__ bb,
                          float* __restrict__ Sc, float* __restrict__ Sh)
{
    const int hI = threadIdx.x;                // 0..127
    const double n = (double)S_ * (double)B_;
    const double mx = mom[0] / n, my = mom[1] / n, mz = mom[2] / n;
    const double cxx = mom[3] / n - mx * mx;
    const double cxy = mom[4] / n - mx * my;
    const double cxz = mom[5] / n - mx * mz;
    const double cyy = mom[6] / n - my * my;
    const double cyz = mom[7] / n - my * mz;
    const double czz = mom[8] / n - mz * mz;
    const double w0 = Wo[hI * 3 + 0], w1 = Wo[hI * 3 + 1], w2 = Wo[hI * 3 + 2];
    const double var = w0 * w0 * cxx + w1 * w1 * cyy + w2 * w2 * czz
                     + 2.0 * (w0 * w1 * cxy + w0 * w2 * cxz + w1 * w2 * cyz);
    const double scale = (double)gamma[hI] / sqrt(var + (double)EPS_);
    const double dotm  = w0 * mx + w1 * my + w2 * mz;
    Sc[hI] = (float)scale;
    Sh[hI] = (float)((double)bb[hI] - scale * dotm);
}

// ---------------------------------------------------------------------------
// Kernel 4: out[b,h,s] = Sc[h] * (Wo_h . states[s,b]) + Sh[h] via
// V_WMMA_F32_16X16X4_F32, fused BN, single 128MB write pass.
// Block = (b, 1024-s chunk); 8 waves x one 16-h tile each.
// States staged in LDS padded to 4 floats/s (4th = 0) so every B fragment is
// one unconditional, 8B-aligned ds_load_b64 — no divergence, no exec masking.
// ---------------------------------------------------------------------------
__global__ __launch_bounds__(256) void k_out(const float* __restrict__ st,
                                             const float* __restrict__ Wo,
                                             const float* __restrict__ Sc,
                                             const float* __restrict__ Sh,
                                             float* __restrict__ out)
{
    __shared__ float lst[1024 * 4];            // 16 KB, padded (k=3 slot is 0)
    const int b = blockIdx.x >> 2;
    const int sbase = (blockIdx.x & 3) << 10;

    for (int i = threadIdx.x; i < 1024; i += 256) {
        const float* sp = st + ((size_t)(sbase + i) * B_ + b) * 3;
        lst[i * 4 + 0] = sp[0];
        lst[i * 4 + 1] = sp[1];
        lst[i * 4 + 2] = sp[2];
        lst[i * 4 + 3] = 0.0f;                 // K=3 zero pad
    }
    __syncthreads();

    const int wave = threadIdx.x >> 5;
    const int lane = threadIdx.x & 31;
    const int lo = lane & 15;
    const int hi = lane >> 4;                  // 0: K={0,1}, rows M=0..7 ; 1: K={2,3}, rows M=8..15
    const int koff = hi << 1;                  // 0 or 2
    const int h0 = wave << 4;                  // 16-h tile per wave
    const int hoff = h0 + (hi << 3);

    // A fragment: 16x4 (M=h, K=k, k=3 zero-padded); one hoisted select only
    v2f a;
    a.x = Wo[(h0 + lo) * 3 + koff];
    a.y = hi ? 0.0f : Wo[(h0 + lo) * 3 + 1];

    // per-row scale/shift (s-invariant, hoisted)
    float scv[8], shv[8];
#pragma unroll
    for (int r = 0; r < 8; ++r) { scv[r] = Sc[hoff + r]; shv[r] = Sh[hoff + r]; }

    float* ob = out + ((size_t)b * H_ + hoff) * S_ + sbase + lo;

#pragma unroll 2
    for (int t = 0; t < 64; ++t) {             // 64 s-tiles of 16 in this chunk
        // B fragment: 4x16 (K=k, N=s); single ds_load_b64, no divergence
        const v2f bf = *(const v2f*)&lst[(((t << 4) + lo) << 2) + koff];
        v8f c = {};
        v8f d = __builtin_amdgcn_wmma_f32_16x16x4_f32(
            /*neg_a=*/false, a, /*neg_b=*/false, bf,
            /*c_mod=*/(short)0, c, /*reuse_a=*/false, /*reuse_b=*/false);
#pragma unroll
        for (int r = 0; r < 8; ++r)
            ob[(size_t)r * S_ + (t << 4)] = fmaf(scv[r], d[r], shv[r]);
    }
}

// ---------------------------------------------------------------------------
extern "C" void kernel_launch(void* const* d_in, const int* in_sizes, int n_in,
                              void* d_out, int out_size, void* d_ws, size_t ws_size,
                              hipStream_t stream) {
    const float* x     = (const float*)d_in[0];
    const float* sig   = (const float*)d_in[1];
    const float* rho   = (const float*)d_in[2];
    const float* bet   = (const float*)d_in[3];
    const float* Wi    = (const float*)d_in[4];
    const float* bi    = (const float*)d_in[5];
    const float* Cm    = (const float*)d_in[6];
    const float* Wo    = (const float*)d_in[7];
    const float* gamma = (const float*)d_in[9];
    const float* bb    = (const float*)d_in[10];
    (void)in_sizes; (void)n_in; (void)out_size; (void)ws_size;

    char* ws = (char*)d_ws;
    float*  ct  = (float*)(ws + WS_CT);
    float*  st  = (float*)(ws + WS_ST);
    float*  ci0 = (float*)(ws + WS_CI0);
    double* mom = (double*)(ws + WS_MOM);
    float*  Sc  = (float*)(ws + WS_SC);
    float*  Sh  = (float*)(ws + WS_SH);
    float*  out = (float*)d_out;

    k_ct     <<<1024, 256, 0, stream>>>(x, Wi, bi, Cm, ct, ci0);
    k_scan   <<<1,     64, 0, stream>>>(ct, ci0, sig, rho, bet, st, mom);
    k_bnparam<<<1,    128, 0, stream>>>(mom, Wo, gamma, bb, Sc, Sh);
    k_out    <<<256,  256, 0, stream>>>(st, Wo, Sc, Sh, out);
}